// GNNModel_81484119539836
// MI455X (gfx1250) — compile-verified
//
#include <hip/hip_runtime.h>

// ---------------------------------------------------------------------------
// 2-layer GCN (Fin=1 -> 16 -> 1) on 250K nodes / 5M edges, gfx1250.
// Scalar edge aggregations (native global_atomic_add via inline asm)
// + a WMMA 16x16x4 F32 per-node MLP.
// ---------------------------------------------------------------------------

typedef float v2f __attribute__((ext_vector_type(2)));
typedef float v8f __attribute__((ext_vector_type(8)));

// Native no-return fp32 atomic add at device scope (RMW in L2, coherent across
// WGPs). Forces GLOBAL_ATOMIC_ADD_F32 instead of a possible CAS-loop expansion.
static __device__ __forceinline__ void atomAddF(float* p, float v) {
    asm volatile("global_atomic_add_f32 %0, %1, off scope:SCOPE_DEV"
                 :
                 : "v"(p), "v"(v)
                 : "memory");
}

// Native u32 atomic add (always a single hardware instruction).
static __device__ __forceinline__ void atomAddU(unsigned* p, unsigned v) {
    __hip_atomic_fetch_add(p, v, __ATOMIC_RELAXED, __HIP_MEMORY_SCOPE_AGENT);
}

// ---- pass 0: deg[n] = 1 (self-loop) ---------------------------------------
__global__ __launch_bounds__(256)
void gcn_init_deg(unsigned* __restrict__ deg, int N) {
    int n = blockIdx.x * blockDim.x + threadIdx.x;
    if (n < N) deg[n] = 1u;
}

// ---- pass 1: deg[dst] += 1 per edge (4 edges/thread, b128 index loads) ----
__global__ __launch_bounds__(256)
void gcn_deg_edges(const int* __restrict__ dst, unsigned* __restrict__ deg, int E) {
    int base = (blockIdx.x * blockDim.x + threadIdx.x) * 4;
    if (base + 3 < E) {
        int4 d4 = *(const int4*)(dst + base);
        atomAddU(&deg[d4.x], 1u);
        atomAddU(&deg[d4.y], 1u);
        atomAddU(&deg[d4.z], 1u);
        atomAddU(&deg[d4.w], 1u);
    } else {
        for (int e = base; e < E; ++e) atomAddU(&deg[dst[e]], 1u);
    }
}

// ---- pass 2: dinv = rsqrt(deg); agg1[n] = dinv[n]^2 * x[n] (self-loop) ----
__global__ __launch_bounds__(256)
void gcn_dinv_selfloop(const unsigned* __restrict__ deg, float* __restrict__ dinv,
                       const float* __restrict__ x, float* __restrict__ agg1, int N) {
    int n = blockIdx.x * blockDim.x + threadIdx.x;
    if (n < N) {
        float d = (float)deg[n];
        float di = (d > 0.0f) ? __frsqrt_rn(d) : 0.0f;
        dinv[n] = di;
        agg1[n] = di * di * x[n];          // self-loop contribution
    }
}

// ---- pass 3: agg1[dst] += dinv[src]*dinv[dst]*x[src] (4 edges/thread) -----
__global__ __launch_bounds__(256)
void gcn_agg1_edges(const int* __restrict__ src, const int* __restrict__ dst,
                    const float* __restrict__ dinv, const float* __restrict__ x,
                    float* __restrict__ agg1, int E) {
    int base = (blockIdx.x * blockDim.x + threadIdx.x) * 4;
    if (base + 3 < E) {
        int4 s4 = *(const int4*)(src + base);
        int4 d4 = *(const int4*)(dst + base);
        atomAddF(&agg1[d4.x], dinv[s4.x] * dinv[d4.x] * x[s4.x]);
        atomAddF(&agg1[d4.y], dinv[s4.y] * dinv[d4.y] * x[s4.y]);
        atomAddF(&agg1[d4.z], dinv[s4.z] * dinv[d4.z] * x[s4.z]);
        atomAddF(&agg1[d4.w], dinv[s4.w] * dinv[d4.w] * x[s4.w]);
    } else {
        for (int e = base; e < E; ++e) {
            int s = src[e], d = dst[e];
            atomAddF(&agg1[d], dinv[s] * dinv[d] * x[s]);
        }
    }
}

// ---- pass 4: per-node MLP via WMMA ----------------------------------------
// One wave handles a tile of 16 nodes:
//   H = agg1_tile (16x1) * W1 (1x16)        -> v_wmma_f32_16x16x4_f32 rank-1
//   H = relu(H + b1); p = H * W2 (per-col)  -> VALU on accumulator layout
//   t[m] = sum over 16 columns              -> shfl_xor tree within 16 lanes
//   out[m] = b2 + dinv[m]^2 * t[m]          (self-loop init for layer 2)
__global__ __launch_bounds__(256)
void gcn_mlp_wmma(const float* __restrict__ agg1, const float* __restrict__ dinv,
                  const float* __restrict__ W1, const float* __restrict__ b1,
                  const float* __restrict__ W2, const float* __restrict__ b2,
                  float* __restrict__ t, float* __restrict__ out,
                  int N, int nTiles) {
    int wave = (int)((blockIdx.x * blockDim.x + threadIdx.x) >> 5);  // tile id
    int lane = (int)(threadIdx.x & 31);
    if (wave >= nTiles) return;            // wave-uniform: EXEC stays all-ones

    int col  = lane & 15;
    int aidx = wave * 16 + col;
    if (aidx > N - 1) aidx = N - 1;        // clamp (N % 16 == 0 in practice)

    // A(16x4): A[m][0] = agg1[tile*16+m]; K=1..3 zero.
    //   layout: V0 lanes0-15 = K0(M=lane), lanes16-31 = K2; V1 = K1/K3.
    float aval = agg1[aidx];
    v2f A; A.x = (lane < 16) ? aval : 0.0f; A.y = 0.0f;
    // B(4x16): B[0][n] = W1[n]; rows 1..3 zero (K0 row at V0 lanes0-15).
    float wv = W1[col];
    v2f B; B.x = (lane < 16) ? wv : 0.0f; B.y = 0.0f;

    v8f C = {};
    C = __builtin_amdgcn_wmma_f32_16x16x4_f32(
            /*neg_a=*/false, A, /*neg_b=*/false, B,
            /*c_mod=*/(short)0, C, /*reuse_a=*/false, /*reuse_b=*/false);

    // Accumulator: C[j] at lane L holds H[M][Ncol], M = j + 8*(L>>4), Ncol = L&15.
    float bcol = b1[col];
    float w2c  = W2[col];
    float p[8];
#pragma unroll
    for (int j = 0; j < 8; ++j) {
        float h = C[j] + bcol;
        h = h > 0.0f ? h : 0.0f;           // ReLU
        p[j] = h * w2c;                    // weight by W2[col]
    }
    // Reduce across the 16 columns (lanes) of each half-wave.
#pragma unroll
    for (int m = 1; m < 16; m <<= 1) {
#pragma unroll
        for (int j = 0; j < 8; ++j) p[j] += __shfl_xor(p[j], m, 32);
    }
    // Lane 0 holds t for rows 0..7, lane 16 for rows 8..15.
    if (col == 0) {
        int base = wave * 16 + ((lane >> 4) << 3);
        float b2v = b2[0];
#pragma unroll
        for (int j = 0; j < 8; ++j) {
            int node = base + j;
            if (node < N) {
                float tv = p[j];
                t[node] = tv;
                float dv = dinv[node];
                out[node] = b2v + dv * dv * tv;   // bias + self-loop term
            }
        }
    }
}

// ---- pass 5: out[dst] += dinv[src]*dinv[dst]*t[src] (4 edges/thread) ------
__global__ __launch_bounds__(256)
void gcn_agg2_edges(const int* __restrict__ src, const int* __restrict__ dst,
                    const float* __restrict__ dinv, const float* __restrict__ t,
                    float* __restrict__ out, int E) {
    int base = (blockIdx.x * blockDim.x + threadIdx.x) * 4;
    if (base + 3 < E) {
        int4 s4 = *(const int4*)(src + base);
        int4 d4 = *(const int4*)(dst + base);
        atomAddF(&out[d4.x], dinv[s4.x] * dinv[d4.x] * t[s4.x]);
        atomAddF(&out[d4.y], dinv[s4.y] * dinv[d4.y] * t[s4.y]);
        atomAddF(&out[d4.z], dinv[s4.z] * dinv[d4.z] * t[s4.z]);
        atomAddF(&out[d4.w], dinv[s4.w] * dinv[d4.w] * t[s4.w]);
    } else {
        for (int e = base; e < E; ++e) {
            int s = src[e], d = dst[e];
            atomAddF(&out[d], dinv[s] * dinv[d] * t[s]);
        }
    }
}

extern "C" void kernel_launch(void* const* d_in, const int* in_sizes, int n_in,
                              void* d_out, int out_size, void* d_ws, size_t ws_size,
                              hipStream_t stream) {
    (void)n_in; (void)ws_size; (void)out_size;
    const float* x   = (const float*)d_in[0];
    const int*   ei  = (const int*)d_in[1];     // [2, E] int32 flat
    const float* W1  = (const float*)d_in[2];   // 16
    const float* b1  = (const float*)d_in[3];   // 16
    const float* W2  = (const float*)d_in[4];   // 16
    const float* b2  = (const float*)d_in[5];   // 1
    float* out = (float*)d_out;

    const int N = in_sizes[0];                  // 250000
    const int E = in_sizes[1] / 2;              // 5000000
    const int* src = ei;
    const int* dst = ei + E;

    // workspace partition: [deg(u32) | dinv | agg1 | t], each N elements
    unsigned* deg  = (unsigned*)d_ws;
    float*    dinv = (float*)(deg + N);
    float*    agg1 = dinv + N;
    float*    tbuf = agg1 + N;

    const int TPB = 256;
    int nBlocksN  = (N + TPB - 1) / TPB;
    int nBlocksE4 = ((E + 3) / 4 + TPB - 1) / TPB;   // 4 edges per thread
    int nTiles    = (N + 15) / 16;
    int nBlocksT  = (nTiles + 7) / 8;                // 8 waves (tiles) per block

    gcn_init_deg<<<nBlocksN, TPB, 0, stream>>>(deg, N);
    gcn_deg_edges<<<nBlocksE4, TPB, 0, stream>>>(dst, deg, E);
    gcn_dinv_selfloop<<<nBlocksN, TPB, 0, stream>>>(deg, dinv, x, agg1, N);
    gcn_agg1_edges<<<nBlocksE4, TPB, 0, stream>>>(src, dst, dinv, x, agg1, E);
    gcn_mlp_wmma<<<nBlocksT, TPB, 0, stream>>>(agg1, dinv, W1, b1, W2, b2,
                                               tbuf, out, N, nTiles);
    gcn_agg2_edges<<<nBlocksE4, TPB, 0, stream>>>(src, dst, dinv, tbuf, out, E);
}